// WildcatPool2d_6794638262969
// MI455X (gfx1250) — compile-verified
//
#include <hip/hip_runtime.h>
#include <hip/hip_bf16.h>

#define WC_BLOCK 256
#define WC_N 4096            // h*w = 64*64
#define WC_K 819             // round(0.2 * 4096)
#define WC_ALPHA 0.7f
#define WC_EPT 16            // elements per thread (4096 / 256)
#define WC_G 4               // slices processed per block
#define WC_SLICES (32 * 512) // b * c

// ---- order-preserving float <-> uint32 mapping (ascending) ----
__device__ __forceinline__ uint32_t wc_f2u(float f) {
  uint32_t b = __float_as_uint(f);
  return b ^ (0x80000000u | (uint32_t)((int32_t)b >> 31));
}
__device__ __forceinline__ float wc_u2f(uint32_t u) {
  uint32_t b = (u & 0x80000000u) ? (u ^ 0x80000000u) : ~u;
  return __uint_as_float(b);
}

// ---- CDNA5 async global -> LDS staging of one 16KB slice (4 x b128 per lane) ----
__device__ __forceinline__ void wc_async_slice(const float* src, const float* dst_lds, int tid) {
  // generic pointer to __shared__ : low 32 bits are the LDS byte offset (shared aperture)
  uint32_t lds0 = (uint32_t)(uintptr_t)dst_lds;
  uint64_t gbase = (uint64_t)(uintptr_t)src;
#pragma unroll
  for (int c = 0; c < 4; ++c) {
    uint32_t voff = (uint32_t)(tid * 16 + c * 4096); // byte offset within slice
    uint32_t ldso = lds0 + voff;
    // GVS mode: vdst = LDS byte address, vaddr = 32-bit offset, saddr = 64-bit base
    asm volatile("global_load_async_to_lds_b128 %0, %1, %2"
                 :: "v"(ldso), "v"(voff), "s"(gbase)
                 : "memory");
  }
}

// ---- block-wide inclusive SUFFIX scan of 256 ints (wave32 shfl + cross-wave), 1 barrier ----
__device__ __forceinline__ int wc_scan_suffix(int v, int tid, volatile int* wtot) {
  const int lane = tid & 31, w = tid >> 5;
#pragma unroll
  for (int off = 1; off < 32; off <<= 1) {
    int o = __shfl_down(v, off, 32);
    if (lane + off < 32) v += o;
  }
  if (lane == 0) wtot[w] = v; // lane 0 holds full-wave sum for suffix scan
  __syncthreads();
  int add = 0;
#pragma unroll
  for (int ww = 0; ww < 8; ++ww) add += (ww > w) ? wtot[ww] : 0;
  return v + add;
}

// ---- block-wide inclusive PREFIX scan of 256 ints ----
__device__ __forceinline__ int wc_scan_prefix(int v, int tid, volatile int* wtot) {
  const int lane = tid & 31, w = tid >> 5;
#pragma unroll
  for (int off = 1; off < 32; off <<= 1) {
    int o = __shfl_up(v, off, 32);
    if (lane >= off) v += o;
  }
  if (lane == 31) wtot[w] = v; // lane 31 holds full-wave sum for prefix scan
  __syncthreads();
  int add = 0;
#pragma unroll
  for (int ww = 0; ww < 8; ++ww) add += (ww < w) ? wtot[ww] : 0;
  return v + add;
}

__global__ __launch_bounds__(WC_BLOCK) void wildcat_pool_kernel(const float* __restrict__ x,
                                                                float* __restrict__ out) {
  __shared__ __align__(16) float buf[2][WC_N]; // double-buffered slice staging (32 KB)
  __shared__ int hist[256];
  __shared__ int sscan[256];
  __shared__ int wtot[8];
  __shared__ int sel;
  __shared__ float wsumT[8], wsumB[8];

  const int tid = threadIdx.x;
  const int slice0 = blockIdx.x * WC_G;

  // prologue: prefetch slices 0 and 1 (8 async b128 ops / thread outstanding)
  wc_async_slice(x + (size_t)(slice0 + 0) * WC_N, buf[0], tid);
  wc_async_slice(x + (size_t)(slice0 + 1) * WC_N, buf[1], tid);

  for (int s = 0; s < WC_G; ++s) {
    const int cur = s & 1;
    // async loads complete in order: <=4 outstanding means slice s's 4 ops are done
    if (s + 1 < WC_G) asm volatile("s_wait_asynccnt 4" ::: "memory");
    else              asm volatile("s_wait_asynccnt 0" ::: "memory");
    __syncthreads(); // all lanes' async data visible

    // copy my 16 elements to registers as sortable keys
    uint32_t uv[WC_EPT];
#pragma unroll
    for (int c = 0; c < 4; ++c) {
      float4 v = reinterpret_cast<const float4*>(buf[cur])[c * 256 + tid];
      uv[4 * c + 0] = wc_f2u(v.x);
      uv[4 * c + 1] = wc_f2u(v.y);
      uv[4 * c + 2] = wc_f2u(v.z);
      uv[4 * c + 3] = wc_f2u(v.w);
    }
    __syncthreads(); // everyone done reading buf[cur] -> safe to overwrite

    // prefetch slice s+2 into the buffer we just drained
    if (s + 2 < WC_G)
      wc_async_slice(x + (size_t)(slice0 + s + 2) * WC_N, buf[cur], tid);

    // ======== k-th LARGEST threshold via 8-bit radix select (4 passes) ========
    int krem = WC_K;
    uint32_t prefix = 0;
    for (int pass = 0; pass < 4; ++pass) {
      hist[tid] = 0;
      __syncthreads();
      if (pass == 0) {
#pragma unroll
        for (int e = 0; e < WC_EPT; ++e) atomicAdd(&hist[uv[e] >> 24], 1);
      } else {
        const int ps = 32 - 8 * pass;
#pragma unroll
        for (int e = 0; e < WC_EPT; ++e)
          if ((uv[e] >> ps) == prefix) atomicAdd(&hist[(uv[e] >> (ps - 8)) & 0xFFu], 1);
      }
      __syncthreads();
      int v = wc_scan_suffix(hist[tid], tid, wtot); // suffix[d] = #elems with digit >= d
      sscan[tid] = v;
      __syncthreads();
      const int nxt = (tid < 255) ? sscan[tid + 1] : 0;
      if (v >= krem && nxt < krem) sel = tid; // exactly one digit satisfies
      __syncthreads();
      const int d = sel;
      const int cnt_above = (d < 255) ? sscan[d + 1] : 0;
      __syncthreads(); // protect sel/sscan/wtot reuse
      krem -= cnt_above;
      prefix = (prefix << 8) | (uint32_t)d;
    }
    const uint32_t thrT = prefix; // key of the K-th largest element
    const int tieT = krem;        // how many threshold-valued elems belong to top-K

    // ======== k-th SMALLEST threshold ========
    krem = WC_K;
    prefix = 0;
    for (int pass = 0; pass < 4; ++pass) {
      hist[tid] = 0;
      __syncthreads();
      if (pass == 0) {
#pragma unroll
        for (int e = 0; e < WC_EPT; ++e) atomicAdd(&hist[uv[e] >> 24], 1);
      } else {
        const int ps = 32 - 8 * pass;
#pragma unroll
        for (int e = 0; e < WC_EPT; ++e)
          if ((uv[e] >> ps) == prefix) atomicAdd(&hist[(uv[e] >> (ps - 8)) & 0xFFu], 1);
      }
      __syncthreads();
      int v = wc_scan_prefix(hist[tid], tid, wtot); // pre[d] = #elems with digit <= d
      sscan[tid] = v;
      __syncthreads();
      const int prv = (tid > 0) ? sscan[tid - 1] : 0;
      if (v >= krem && prv < krem) sel = tid;
      __syncthreads();
      const int d = sel;
      const int cnt_below = (d > 0) ? sscan[d - 1] : 0;
      __syncthreads();
      krem -= cnt_below;
      prefix = (prefix << 8) | (uint32_t)d;
    }
    const uint32_t thrB = prefix; // key of the K-th smallest element
    const int tieB = krem;

    // ======== exact sums (strictly beyond threshold + tie correction) ========
    float sT = 0.f, sB = 0.f;
#pragma unroll
    for (int e = 0; e < WC_EPT; ++e) {
      const float xe = wc_u2f(uv[e]);
      if (uv[e] > thrT) sT += xe;
      if (uv[e] < thrB) sB += xe;
    }
#pragma unroll
    for (int off = 16; off > 0; off >>= 1) {
      sT += __shfl_xor(sT, off, 32);
      sB += __shfl_xor(sB, off, 32);
    }
    if ((tid & 31) == 0) { wsumT[tid >> 5] = sT; wsumB[tid >> 5] = sB; }
    __syncthreads();
    if (tid == 0) {
      float tT = 0.f, tB = 0.f;
      for (int w = 0; w < 8; ++w) { tT += wsumT[w]; tB += wsumB[w]; }
      tT += (float)tieT * wc_u2f(thrT);
      tB += (float)tieB * wc_u2f(thrB);
      out[slice0 + s] = tT * (1.0f / WC_K) + (WC_ALPHA / WC_K) * tB;
    }
    __syncthreads(); // protect wsum/hist reuse across slice iterations
  }
}

extern "C" void kernel_launch(void* const* d_in, const int* in_sizes, int n_in,
                              void* d_out, int out_size, void* d_ws, size_t ws_size,
                              hipStream_t stream) {
  (void)in_sizes; (void)n_in; (void)d_ws; (void)ws_size; (void)out_size;
  const float* x = (const float*)d_in[0]; // (32, 512, 64, 64) fp32
  float* out = (float*)d_out;             // (32, 512) fp32
  const int blocks = WC_SLICES / WC_G;    // 4096 blocks of 256 threads
  wildcat_pool_kernel<<<blocks, WC_BLOCK, 0, stream>>>(x, out);
}